// SymmetricContraction_17600775979387
// MI455X (gfx1250) — compile-verified
//
#include <hip/hip_runtime.h>

// ---------------------------------------------------------------------------
// Symmetric contraction (MACE-style), B=1024 C=256 ELL=9 E=10, K1/K2/K3=1/4/20
// Strategy: fold everything into two tall-skinny fp32 GEMMs executed with
// V_WMMA_F32_16X16X4_F32 (wave32, M=16,N=16,K=4), K padded to 184 (U3:180 + U2:4).
// ---------------------------------------------------------------------------

typedef __attribute__((ext_vector_type(2))) float v2f;
typedef __attribute__((ext_vector_type(8))) float v8f;

#define NKP   184          // padded GEMM-K  (180 = 9*20 U3 terms, +4 U2 terms)
#define NP0   96           // padded N for l0 (81 -> 96)
#define NP1   256          // padded N for l1 (243 -> 256)
#define NROWS (1024*256)   // B*C rows
#define OUT1_OFF NROWS

// --------------------------- pack kernel -----------------------------------
// Build N-major B matrices in workspace:
//   Bp0[n][k] : n in [0,96),  k in [0,184)   (l=0;  U3_l0 flat is n*180+k)
//   Bp1[n][k] : n in [0,256), k in [0,184)   (l=1;  U3_l1 flat is n*180+k)
// k in [180,184) holds U2 columns; padded n rows are zero.
__global__ void symcon_pack(const float* __restrict__ U2_l0,
                            const float* __restrict__ U3_l0,
                            const float* __restrict__ U2_l1,
                            const float* __restrict__ U3_l1,
                            float* __restrict__ Bp0,
                            float* __restrict__ Bp1) {
  int idx = blockIdx.x * 256 + threadIdx.x;
  if (idx < NP0 * NKP) {
    int n = idx / NKP, k = idx % NKP;
    float v = 0.f;
    if (n < 81) v = (k < 180) ? U3_l0[n * 180 + k] : U2_l0[n * 4 + (k - 180)];
    Bp0[idx] = v;
  } else {
    idx -= NP0 * NKP;
    if (idx < NP1 * NKP) {
      int n = idx / NKP, k = idx % NKP;
      float v = 0.f;
      if (n < 243) v = (k < 180) ? U3_l1[n * 180 + k] : U2_l1[n * 4 + (k - 180)];
      Bp1[idx] = v;
    }
  }
}

// --------------------------- device helpers --------------------------------

// Z[r,k'] : k'<180 -> x[k'/20] * P3[k'%20] ; else P2[k'-180].
// Only called with compile-time k (loops fully unrolled) -> register indices.
__device__ __forceinline__ float zval(int k, const float (&xr)[9],
                                      const float (&p3)[20], const float (&p2)[4]) {
  return (k < 180) ? xr[k / 20] * p3[k % 20] : p2[k - 180];
}

// Per-row weight projections: P = sum_e y[b,e] * w[e,k,c].
__device__ __forceinline__ void compute_p(const float (&yv)[10], int c,
    const float* __restrict__ w1, const float* __restrict__ w2,
    const float* __restrict__ w3,
    float (&p3)[20], float (&p2)[4], float &p1) {
  #pragma unroll
  for (int k = 0; k < 20; ++k) p3[k] = 0.f;
  #pragma unroll
  for (int k = 0; k < 4; ++k) p2[k] = 0.f;
  p1 = 0.f;
  #pragma unroll
  for (int e = 0; e < 10; ++e) {
    float ye = yv[e];
    #pragma unroll
    for (int k = 0; k < 20; ++k) p3[k] = fmaf(ye, w3[(e * 20 + k) * 256 + c], p3[k]);
    #pragma unroll
    for (int k = 0; k < 4; ++k)  p2[k] = fmaf(ye, w2[(e * 4 + k) * 256 + c], p2[k]);
    p1 = fmaf(ye, w1[e * 256 + c], p1);
  }
}

// One GEMM segment: NT adjacent 16-col tiles, K = 46 wmma-steps of 4.
// A fragment: lane(l) -> M = l&15; VGPR0/1 hold K = {0,1}+2*(l>=16).
// B fragment (N-major storage): single b64 load per (s,t):
//   addr = Bp + ((n0 + t*16 + (l&15))*184 + 4s + 2*(l>=16)) * 4
__device__ __forceinline__ void gemm_seg(const float* __restrict__ bseg, int n0,
    int NT8, int lan, bool hi,
    const float (&xr)[9], const float (&p3)[20], const float (&p2)[4],
    v8f* __restrict__ acc, int NT) {
  (void)NT8;
  const char* bp = (const char*)bseg +
                   (size_t)(((n0 + lan) * NKP + (hi ? 2 : 0)) * 4);
  #pragma unroll
  for (int s = 0; s < 46; ++s) {
    float z0 = zval(4 * s + 0, xr, p3, p2);
    float z1 = zval(4 * s + 1, xr, p3, p2);
    float z2 = zval(4 * s + 2, xr, p3, p2);
    float z3 = zval(4 * s + 3, xr, p3, p2);
    v2f a;
    a.x = hi ? z2 : z0;
    a.y = hi ? z3 : z1;
    #pragma unroll
    for (int t = 0; t < 8; ++t) {
      if (t < NT) {
        v2f bv = *(const v2f*)(bp + t * (16 * NKP * 4) + s * 16);
        acc[t] = __builtin_amdgcn_wmma_f32_16x16x4_f32(
            false, a, false, bv, (short)0, acc[t], false, false);
      }
    }
  }
}

// --------------------------- main kernel -----------------------------------
__global__ __launch_bounds__(256, 1) void symcon_main(
    const float* __restrict__ x, const float* __restrict__ y,
    const float* __restrict__ U1_l0, const float* __restrict__ U1_l1,
    const float* __restrict__ w1_l0, const float* __restrict__ w2_l0,
    const float* __restrict__ w3_l0,
    const float* __restrict__ w1_l1, const float* __restrict__ w2_l1,
    const float* __restrict__ w3_l1,
    const float* __restrict__ Bp0, const float* __restrict__ Bp1,
    float* __restrict__ out) {
  // per-wave staging of the strip's x rows (+ P1 values) for the epilogue
  __shared__ float xs[8][16][12];   // [wave][row][ x0..x8, p1_l0, p1_l1, pad ]

  const int tid  = threadIdx.x;
  const int wv   = tid >> 5;
  const int lane = tid & 31;
  const int lan  = lane & 15;
  const bool hi  = lane >= 16;
  const int row0 = (blockIdx.x * 8 + wv) * 16;   // strip base row
  const int row  = row0 + lan;                   // this lane's A-matrix row
  const int b    = row >> 8;
  const int c    = row & 255;

  float xr[9];
  #pragma unroll
  for (int i = 0; i < 9; ++i) xr[i] = x[row * 9 + i];

  float yv[10];
  #pragma unroll
  for (int e = 0; e < 10; ++e) yv[e] = y[b * 10 + e];

  float p3[20], p2[4], p1;

  // ================= l = 0 =================
  compute_p(yv, c, w1_l0, w2_l0, w3_l0, p3, p2, p1);
  if (!hi) {
    #pragma unroll
    for (int i = 0; i < 9; ++i) xs[wv][lan][i] = xr[i];
    xs[wv][lan][9] = p1;
  }

  {
    v8f acc6[6];
    v8f vz = {};
    #pragma unroll
    for (int t = 0; t < 6; ++t) acc6[t] = vz;

    gemm_seg(Bp0, 0, 0, lan, hi, xr, p3, p2, acc6, 6);

    // epilogue: out0[r] = sum_{n=wx<81} c2[r,n] * x[r,w]*x[r,x]
    float o0[8];
    #pragma unroll
    for (int j = 0; j < 8; ++j) o0[j] = 0.f;
    #pragma unroll
    for (int t = 0; t < 6; ++t) {
      int n = t * 16 + lan;
      if (n < 81) {
        int wq = n / 9, xq = n % 9;
        #pragma unroll
        for (int j = 0; j < 8; ++j) {
          int rl = j + (hi ? 8 : 0);               // D-row for VGPR j
          float q = xs[wv][rl][wq] * xs[wv][rl][xq];
          o0[j] = fmaf(acc6[t][j], q, o0[j]);
        }
      }
    }
    // reduce over the 16 lanes of each half-wave (N dimension)
    #pragma unroll
    for (int j = 0; j < 8; ++j) {
      o0[j] += __shfl_xor(o0[j], 1, 32);
      o0[j] += __shfl_xor(o0[j], 2, 32);
      o0[j] += __shfl_xor(o0[j], 4, 32);
      o0[j] += __shfl_xor(o0[j], 8, 32);
    }
    if (lan == 0) {                                 // lanes 0 and 16 write
      #pragma unroll
      for (int j = 0; j < 8; ++j) {
        int rl = j + (hi ? 8 : 0);
        float du = 0.f;
        #pragma unroll
        for (int wq = 0; wq < 9; ++wq) du = fmaf(U1_l0[wq], xs[wv][rl][wq], du);
        out[row0 + rl] = o0[j] + xs[wv][rl][9] * du;
      }
    }
  }

  // ================= l = 1 =================
  compute_p(yv, c, w1_l1, w2_l1, w3_l1, p3, p2, p1);
  if (!hi) xs[wv][lan][10] = p1;

  float o1[8][3];
  #pragma unroll
  for (int j = 0; j < 8; ++j)
    #pragma unroll
    for (int wq = 0; wq < 3; ++wq) o1[j][wq] = 0.f;

  v8f acc8[8];
  #pragma unroll
  for (int seg = 0; seg < 2; ++seg) {
    const int n0 = seg * 128;
    v8f vz = {};
    #pragma unroll
    for (int t = 0; t < 8; ++t) acc8[t] = vz;

    gemm_seg(Bp1, n0, 0, lan, hi, xr, p3, p2, acc8, 8);

    #pragma unroll
    for (int t = 0; t < 8; ++t) {
      int n = n0 + t * 16 + lan;
      if (n < 243) {
        int wq = n / 81, xq = (n / 9) % 9, vq = n % 9;
        #pragma unroll
        for (int j = 0; j < 8; ++j) {
          int rl = j + (hi ? 8 : 0);
          float q = xs[wv][rl][xq] * xs[wv][rl][vq];
          float pz = acc8[t][j] * q;
          if (wq == 0)      o1[j][0] += pz;
          else if (wq == 1) o1[j][1] += pz;
          else              o1[j][2] += pz;
        }
      }
    }
  }

  #pragma unroll
  for (int j = 0; j < 8; ++j)
    #pragma unroll
    for (int wq = 0; wq < 3; ++wq) {
      float v = o1[j][wq];
      v += __shfl_xor(v, 1, 32);
      v += __shfl_xor(v, 2, 32);
      v += __shfl_xor(v, 4, 32);
      v += __shfl_xor(v, 8, 32);
      o1[j][wq] = v;
    }
  if (lan == 0) {
    #pragma unroll
    for (int j = 0; j < 8; ++j) {
      int rl = j + (hi ? 8 : 0);
      int r  = row0 + rl;
      #pragma unroll
      for (int wq = 0; wq < 3; ++wq) {
        float du = 0.f;
        #pragma unroll
        for (int xq = 0; xq < 9; ++xq)
          du = fmaf(U1_l1[wq * 9 + xq], xs[wv][rl][xq], du);
        out[OUT1_OFF + r * 3 + wq] = o1[j][wq] + xs[wv][rl][10] * du;
      }
    }
  }
}

// --------------------------- host launch -----------------------------------
extern "C" void kernel_launch(void* const* d_in, const int* in_sizes, int n_in,
                              void* d_out, int out_size, void* d_ws, size_t ws_size,
                              hipStream_t stream) {
  (void)in_sizes; (void)n_in; (void)out_size; (void)ws_size;
  const float* x     = (const float*)d_in[0];
  const float* y     = (const float*)d_in[1];
  const float* U1_l0 = (const float*)d_in[2];
  const float* U2_l0 = (const float*)d_in[3];
  const float* U3_l0 = (const float*)d_in[4];
  const float* U1_l1 = (const float*)d_in[5];
  const float* U2_l1 = (const float*)d_in[6];
  const float* U3_l1 = (const float*)d_in[7];
  const float* w1_l0 = (const float*)d_in[8];
  const float* w2_l0 = (const float*)d_in[9];
  const float* w3_l0 = (const float*)d_in[10];
  const float* w1_l1 = (const float*)d_in[11];
  const float* w2_l1 = (const float*)d_in[12];
  const float* w3_l1 = (const float*)d_in[13];

  float* Bp0 = (float*)d_ws;
  float* Bp1 = Bp0 + NP0 * NKP;

  const int pack_elems = (NP0 + NP1) * NKP;
  symcon_pack<<<(pack_elems + 255) / 256, 256, 0, stream>>>(
      U2_l0, U3_l0, U2_l1, U3_l1, Bp0, Bp1);

  // 262144 rows / (8 waves * 16 rows) = 2048 workgroups
  symcon_main<<<2048, 256, 0, stream>>>(
      x, y, U1_l0, U1_l1,
      w1_l0, w2_l0, w3_l0,
      w1_l1, w2_l1, w3_l1,
      Bp0, Bp1, (float*)d_out);
}